// LocMotionAppearance_62955630625216
// MI455X (gfx1250) — compile-verified
//
#include <hip/hip_runtime.h>

#define NB   2
#define HH   512
#define WW   512
#define NSPX 1024
#define NSEG (NB * NSPX)     // 2048 superpixels
#define NPIX (NB * HH * WW)  // 524288 pixels
#define NE   32768           // edges
#define PCH  228             // pooled channels: [2 coord |1 fx |1 fy |32 s0 |64 s1 |128 s2]

typedef __attribute__((ext_vector_type(2))) float v2f;
typedef __attribute__((ext_vector_type(8))) float v8f;

// ---------------------------------------------------------------- utilities
__global__ void zero_kernel(float* p, int n) {
  int i = blockIdx.x * blockDim.x + threadIdx.x;
  if (i < n) p[i] = 0.0f;
}

__global__ void copy_cols_kernel(const float* __restrict__ src, int ldsrc,
                                 float* __restrict__ dst, int lddst,
                                 int C, int nrows) {
  int i = blockIdx.x * blockDim.x + threadIdx.x;
  int total = nrows * C;
  if (i >= total) return;
  int r = i / C, c = i - r * C;
  dst[(size_t)r * lddst + c] = src[(size_t)r * ldsrc + c];
}

// X[r, c] /= max(den[r], 1)   (X is nrows x C contiguous)
__global__ void rowdiv_kernel(float* __restrict__ X, const float* __restrict__ den,
                              int C, int total) {
  int i = blockIdx.x * blockDim.x + threadIdx.x;
  if (i >= total) return;
  int r = i / C;
  X[i] /= fmaxf(den[r], 1.0f);
}

// ------------------------------------------------------- superpixel pooling
__global__ void pool_kernel(const int* __restrict__ labels,
                            const float* __restrict__ fx, const float* __restrict__ fy,
                            const float* __restrict__ s0, const float* __restrict__ s1,
                            const float* __restrict__ s2,
                            float* __restrict__ P, float* __restrict__ cnt) {
  int idx = blockIdx.x * blockDim.x + threadIdx.x;
  if (idx >= NPIX) return;
  const int HWp = HH * WW;
  int b  = idx / HWp;
  int hw = idx - b * HWp;
  int h  = hw / WW;
  int w  = hw - h * WW;
  int seg = labels[idx] + b * NSPX;
  float* pr = P + (size_t)seg * PCH;
  atomicAdd(&cnt[seg], 1.0f);
  atomicAdd(pr + 0, (float)h * (1.0f / (float)(HH - 1)));
  atomicAdd(pr + 1, (float)w * (1.0f / (float)(WW - 1)));
  atomicAdd(pr + 2, fx[idx]);
  atomicAdd(pr + 3, fy[idx]);
#pragma unroll 4
  for (int c = 0; c < 32; ++c)
    atomicAdd(pr + 4 + c, s0[((size_t)(b * 32 + c)) * HWp + hw]);
#pragma unroll 4
  for (int c = 0; c < 64; ++c)
    atomicAdd(pr + 36 + c, s1[((size_t)(b * 64 + c)) * HWp + hw]);
#pragma unroll 4
  for (int c = 0; c < 128; ++c)
    atomicAdd(pr + 100 + c, s2[((size_t)(b * 128 + c)) * HWp + hw]);
}

// --------------------------------------------------------------- edge stuff
__global__ void edge_den_kernel(const int* __restrict__ edges,
                                float* __restrict__ denp, float* __restrict__ denn) {
  int e = blockIdx.x * blockDim.x + threadIdx.x;
  if (e >= NE) return;
  int tgt = edges[NE + e];
  int sgn = edges[2 * NE + e];
  if (sgn == -1) atomicAdd(&denn[tgt], 1.0f);
  else           atomicAdd(&denp[tgt], 1.0f);
}

// num[tgt, c] += X[src, c] for edges whose sign matches (wantneg: sign==-1)
__global__ void edge_scatter_kernel(const float* __restrict__ X, int ldx,
                                    const int* __restrict__ edges, int wantneg,
                                    float* __restrict__ num, int C) {
  long total  = (long)NE * C;
  long stride = (long)gridDim.x * blockDim.x;
  for (long t = (long)blockIdx.x * blockDim.x + threadIdx.x; t < total; t += stride) {
    int e = (int)(t / C);
    int c = (int)(t - (long)e * C);
    int sgn = edges[2 * NE + e];
    if ((sgn == -1) != (wantneg != 0)) continue;
    int s = edges[e];
    int d = edges[NE + e];
    atomicAdd(&num[(size_t)d * C + c], X[(size_t)s * ldx + c]);
  }
}

// ----------------------------------------------- batchnorm (+ReLU) per column
__global__ void bn_relu_kernel(const float* __restrict__ in, int ldin,
                               float* __restrict__ out, int ldout,
                               const float* __restrict__ gamma,
                               const float* __restrict__ beta, int nrows) {
  int c   = blockIdx.x;
  int tid = threadIdx.x;
  __shared__ float ssum[256];
  __shared__ float ssq[256];
  float s = 0.0f, q = 0.0f;
  for (int r = tid; r < nrows; r += blockDim.x) {
    float v = in[(size_t)r * ldin + c];
    s += v; q += v * v;
  }
  ssum[tid] = s; ssq[tid] = q;
  __syncthreads();
  for (int off = 128; off > 0; off >>= 1) {
    if (tid < off) { ssum[tid] += ssum[tid + off]; ssq[tid] += ssq[tid + off]; }
    __syncthreads();
  }
  float mean = ssum[0] / (float)nrows;
  float var  = ssq[0] / (float)nrows - mean * mean;
  float inv  = rsqrtf(var + 1e-5f);
  float g = gamma[c], b = beta[c];
  for (int r = tid; r < nrows; r += blockDim.x) {
    float v = in[(size_t)r * ldin + c];
    float y = g * (v - mean) * inv + b;
    out[(size_t)r * ldout + c] = fmaxf(y, 0.0f);
  }
}

// ------------------------------------------------ fp32 WMMA GEMM: Y = X@W^T
// Y[n, m] (+)= sum_k X[n*ldx + k] * W[m*ldw + woff + k]  (+ bias[m]) (+ReLU)
// One wave per 16x16 tile of Y. V_WMMA_F32_16X16X4_F32, K-step 4.
__global__ void gemm_wmma_kernel(float* __restrict__ Y, int ldy,
                                 const float* __restrict__ X, int ldx,
                                 const float* __restrict__ W, int ldw, int woff,
                                 const float* __restrict__ bias,
                                 int Kc, int accflag, int reluflag) {
  int lane = threadIdx.x;                                    // 0..31
  int bm = blockIdx.x * 16;                                  // X-row tile base
  int tn0 = (blockIdx.y * blockDim.y + threadIdx.y) * 16;    // output-col tile base
  int mr = lane & 15;          // A row / B col within tile
  int kh = (lane >> 4) * 2;    // k-pair select (ISA 16x4 / 4x16 fp32 layout)
  const float* xrow = X + (size_t)(bm  + mr) * ldx + kh;
  const float* wrow = W + (size_t)(tn0 + mr) * ldw + woff + kh;

  v8f accv = {0.f, 0.f, 0.f, 0.f, 0.f, 0.f, 0.f, 0.f};
#pragma unroll 4
  for (int k = 0; k < Kc; k += 4) {
    v2f a, b;
    a.x = xrow[k]; a.y = xrow[k + 1];
    b.x = wrow[k]; b.y = wrow[k + 1];
    accv = __builtin_amdgcn_wmma_f32_16x16x4_f32(
        false, a, false, b, (short)0, accv, false, false);
  }

  // C/D layout: VGPR v holds row v (lanes 0-15) / row v+8 (lanes 16-31), col = lane&15
  int cn  = lane & 15;
  int rhi = (lane >> 4) * 8;
  float bv = (bias != nullptr) ? bias[tn0 + cn] : 0.0f;
#pragma unroll
  for (int v = 0; v < 8; ++v) {
    int row = bm + v + rhi;
    float val = accv[v] + bv;
    float* yp = Y + (size_t)row * ldy + tn0 + cn;
    if (accflag)  val += *yp;
    if (reluflag) val = fmaxf(val, 0.0f);
    *yp = val;
  }
}

// ---------------------------------------------------------------- launcher
extern "C" void kernel_launch(void* const* d_in, const int* in_sizes, int n_in,
                              void* d_out, int out_size, void* d_ws, size_t ws_size,
                              hipStream_t stream) {
  (void)in_sizes; (void)n_in; (void)out_size; (void)ws_size;
  const int*   labels   = (const int*)d_in[0];
  const int*   edges    = (const int*)d_in[1];
  const float* fx       = (const float*)d_in[2];
  const float* fy       = (const float*)d_in[3];
  const float* skip0    = (const float*)d_in[4];
  const float* skip1    = (const float*)d_in[5];
  const float* skip2    = (const float*)d_in[6];
  const float* bn_pre_g = (const float*)d_in[7];
  const float* bn_pre_b = (const float*)d_in[8];
  const float* g0_Wpl   = (const float*)d_in[9];
  const float* g0_Wpr   = (const float*)d_in[10];
  const float* g0_bp    = (const float*)d_in[11];
  const float* g0_Wnl   = (const float*)d_in[12];
  const float* g0_Wnr   = (const float*)d_in[13];
  const float* g0_bng   = (const float*)d_in[14];
  const float* g0_gamma = (const float*)d_in[15];
  const float* g0_beta  = (const float*)d_in[16];
  const float* g1_Wp    = (const float*)d_in[17];
  const float* g1_bp    = (const float*)d_in[18];
  const float* g1_Wn    = (const float*)d_in[19];
  const float* g1_bng   = (const float*)d_in[20];
  const float* g1_gamma = (const float*)d_in[21];
  const float* g1_beta  = (const float*)d_in[22];
  const float* g2_Wp    = (const float*)d_in[23];
  const float* g2_bp    = (const float*)d_in[24];
  const float* g2_Wn    = (const float*)d_in[25];
  const float* g2_bng   = (const float*)d_in[26];
  const float* g2_gamma = (const float*)d_in[27];
  const float* g2_beta  = (const float*)d_in[28];
  const float* m1p_W    = (const float*)d_in[29];
  const float* m1p_b    = (const float*)d_in[30];
  const float* m1p_g    = (const float*)d_in[31];
  const float* m1p_bt   = (const float*)d_in[32];
  const float* m1n_W    = (const float*)d_in[33];
  const float* m1n_b    = (const float*)d_in[34];
  const float* m1n_g    = (const float*)d_in[35];
  const float* m1n_bt   = (const float*)d_in[36];
  const float* m2p_W    = (const float*)d_in[37];
  const float* m2p_b    = (const float*)d_in[38];
  const float* m2p_g    = (const float*)d_in[39];
  const float* m2p_bt   = (const float*)d_in[40];
  const float* m2n_W    = (const float*)d_in[41];
  const float* m2n_b    = (const float*)d_in[42];
  const float* m2n_g    = (const float*)d_in[43];
  const float* m2n_bt   = (const float*)d_in[44];
  const float* pw_W     = (const float*)d_in[45];
  const float* pw_b     = (const float*)d_in[46];
  float* out = (float*)d_out;

  // ---- workspace layout (floats) ----
  float* P    = (float*)d_ws;        // NSEG x 228  pooled [coord2|fx|fy|s0|s1|s2]
  float* cnt  = P    + (size_t)NSEG * PCH;   // NSEG
  float* denp = cnt  + NSEG;                 // NSEG
  float* denn = denp + NSEG;                 // NSEG
  float* x0   = denn + NSEG;                 // NSEG x 36
  float* mmA  = x0   + (size_t)NSEG * 36;    // NSEG x 128 (mmean scratch)
  float* mmB  = mmA  + (size_t)NSEG * 128;
  float* mmC  = mmB  + (size_t)NSEG * 128;
  float* mmD  = mmC  + (size_t)NSEG * 128;
  float* opon = mmD  + (size_t)NSEG * 128;   // NSEG x 512 (op|on concat)
  float* x    = opon + (size_t)NSEG * 512;   // NSEG x 512 (current activation)
  float* abuf = x    + (size_t)NSEG * 512;   // NSEG x 128
  float* cbuf = abuf + (size_t)NSEG * 128;   // NSEG x 128
  float* tp   = cbuf + (size_t)NSEG * 128;   // NSEG x 128 (pre-BN)
  float* tn   = tp   + (size_t)NSEG * 128;   // NSEG x 128 (pre-BN)

  auto ZERO = [&](float* p, int n) {
    zero_kernel<<<(n + 255) / 256, 256, 0, stream>>>(p, n);
  };
  auto GEMM = [&](float* Y, int ldy, const float* X_, int ldx,
                  const float* W_, int ldw, int woff, const float* bias,
                  int M, int Kc, int acc, int relu) {
    dim3 grid(NSEG / 16, M / 64), block(32, 4);
    gemm_wmma_kernel<<<grid, block, 0, stream>>>(Y, ldy, X_, ldx, W_, ldw, woff,
                                                 bias, Kc, acc, relu);
  };
  auto BN = [&](const float* in, int ldin, float* o, int ldo,
                const float* ga, const float* be, int C) {
    bn_relu_kernel<<<C, 256, 0, stream>>>(in, ldin, o, ldo, ga, be, NSEG);
  };
  auto MMEAN = [&](const float* Xsrc, int ldx, float* num, int C, int wantneg,
                   const float* den) {
    ZERO(num, NSEG * C);
    long total = (long)NE * C;
    int blocks = (int)((total + 255) / 256);
    edge_scatter_kernel<<<blocks, 256, 0, stream>>>(Xsrc, ldx, edges, wantneg, num, C);
    rowdiv_kernel<<<(NSEG * C + 255) / 256, 256, 0, stream>>>(num, den, C, NSEG * C);
  };

  // ---- stage 1: superpixel mean pooling (bandwidth-bound) ----
  ZERO(P, NSEG * PCH); ZERO(cnt, NSEG); ZERO(denp, NSEG); ZERO(denn, NSEG);
  pool_kernel<<<(NPIX + 255) / 256, 256, 0, stream>>>(labels, fx, fy, skip0, skip1,
                                                      skip2, P, cnt);
  rowdiv_kernel<<<(NSEG * PCH + 255) / 256, 256, 0, stream>>>(P, cnt, PCH, NSEG * PCH);
  edge_den_kernel<<<(NE + 255) / 256, 256, 0, stream>>>(edges, denp, denn);

  // ---- x0 = concat[relu(bn(pooled[0:4])), s0] : NSEG x 36 ----
  BN(P, PCH, x0, 36, bn_pre_g, bn_pre_b, 4);
  copy_cols_kernel<<<(NSEG * 32 + 255) / 256, 256, 0, stream>>>(P + 4, PCH, x0 + 4, 36,
                                                                32, NSEG);

  // ---- g0: op = mmp(x0)@Wpl.T + x0@Wpr.T + bp ; on likewise ----
  MMEAN(x0, 36, mmA, 36, 0, denp);
  MMEAN(x0, 36, mmB, 36, 1, denn);
  GEMM(opon,      128, mmA, 36, g0_Wpl, 36, 0, g0_bp,  64, 36, 0, 0);
  GEMM(opon,      128, x0,  36, g0_Wpr, 36, 0, nullptr, 64, 36, 1, 0);
  GEMM(opon + 64, 128, mmB, 36, g0_Wnl, 36, 0, g0_bng, 64, 36, 0, 0);
  GEMM(opon + 64, 128, x0,  36, g0_Wnr, 36, 0, nullptr, 64, 36, 1, 0);
  BN(opon, 128, x, 128, g0_gamma, g0_beta, 128);   // x : NSEG x 128

  // ---- two graph layers ----
  struct Layer {
    int half; const float* skipP;
    const float *mpW, *mpb, *mpg, *mpbt, *mnW, *mnb, *mng, *mnbt;
    const float *Wp, *bp, *Wn, *bng, *gam, *bet;
  } layers[2] = {
    { 64,  P + 36,  m1p_W, m1p_b, m1p_g, m1p_bt, m1n_W, m1n_b, m1n_g, m1n_bt,
      g1_Wp, g1_bp, g1_Wn, g1_bng, g1_gamma, g1_beta },
    { 128, P + 100, m2p_W, m2p_b, m2p_g, m2p_bt, m2n_W, m2n_b, m2n_g, m2n_bt,
      g2_Wp, g2_bp, g2_Wn, g2_bng, g2_gamma, g2_beta },
  };

  for (int L = 0; L < 2; ++L) {
    const Layer& ly = layers[L];
    int half = ly.half;
    int C = 2 * half;   // width of incoming x
    int M = 2 * half;   // output width of Wp/Wn

    // xp_raw = concat[x[:,half:], skip] @ mpW.T + mpb   (K-chunked, no concat copy)
    GEMM(tp, half, x + half,  C,   ly.mpW, C, 0,    ly.mpb,  half, half, 0, 0);
    GEMM(tp, half, ly.skipP,  PCH, ly.mpW, C, half, nullptr, half, half, 1, 0);
    BN(tp, half, abuf, half, ly.mpg, ly.mpbt, half);           // a = xp

    // xn_raw = concat[x[:,:half], skip] @ mnW.T + mnb
    GEMM(tn, half, x,         C,   ly.mnW, C, 0,    ly.mnb,  half, half, 0, 0);
    GEMM(tn, half, ly.skipP,  PCH, ly.mnW, C, half, nullptr, half, half, 1, 0);
    BN(tn, half, cbuf, half, ly.mng, ly.mnbt, half);           // c = xn

    MMEAN(abuf, half, mmA, half, 0, denp);   // mmp(a)
    MMEAN(cbuf, half, mmB, half, 1, denn);   // mmn(c)
    MMEAN(cbuf, half, mmC, half, 0, denp);   // mmp(c)
    MMEAN(abuf, half, mmD, half, 1, denn);   // mmn(a)

    // op = [mmp(a)|mmn(c)|a] @ Wp.T + bp   -> opon cols [0,M)
    GEMM(opon,     2 * M, mmA,  half, ly.Wp, 3 * half, 0,        ly.bp,  M, half, 0, 0);
    GEMM(opon,     2 * M, mmB,  half, ly.Wp, 3 * half, half,     nullptr, M, half, 1, 0);
    GEMM(opon,     2 * M, abuf, half, ly.Wp, 3 * half, 2 * half, nullptr, M, half, 1, 0);
    // on = [mmp(c)|mmn(a)|c] @ Wn.T + bn   -> opon cols [M,2M)
    GEMM(opon + M, 2 * M, mmC,  half, ly.Wn, 3 * half, 0,        ly.bng, M, half, 0, 0);
    GEMM(opon + M, 2 * M, mmD,  half, ly.Wn, 3 * half, half,     nullptr, M, half, 1, 0);
    GEMM(opon + M, 2 * M, cbuf, half, ly.Wn, 3 * half, 2 * half, nullptr, M, half, 1, 0);

    BN(opon, 2 * M, x, 2 * M, ly.gam, ly.bet, 2 * M);  // x : NSEG x (2M)
  }

  // ---- final pointwise: out = relu(x @ pw_W.T + pw_b), NSEG x 256 ----
  GEMM(out, 256, x, 512, pw_W, 512, 0, pw_b, 256, 512, 0, 1);
}